// SprecherLayerBlock_53257594470587
// MI455X (gfx1250) — compile-verified
//
#include <hip/hip_runtime.h>
#include <stdint.h>

// Problem constants (from the reference)
#define B_TOTAL 8192
#define D_IN    128
#define D_OUT   128
#define PHI_K   100
#define PHIU_K  100
#define ROWS    8          // batch rows per block (ILP + LDS reuse)
#define EPSF    1e-8f

// ws layout (floats):
//   [0..99]    phi_coeffs (normalized cumsum)
//   [100] phi_max [101] inv_h1 [102] dmin [103] inv_h2
//   [104] left_slope [105] right_slope [106] dmax [107] eta
//   [128..327] 100 float2 pairs: (phi[k], phi[k+1]-phi[k]); pair[99]=(phi[99],0)
#define WS_PAIRS 128

// ---------------------------------------------------------------------------
// One-thread prologue (runs on the CDNA5 scalar-float SALU pipe):
// softplus -> cumsum -> normalize; lambda min/max; emit fused lerp pairs
// with a sentinel endpoint pair so the hot loop needs no index clamp.
// ---------------------------------------------------------------------------
__global__ void sprecher_setup(const float* __restrict__ phi_log_inc,
                               const float* __restrict__ Phi_coeffs,
                               const float* __restrict__ lambdas,
                               const float* __restrict__ eta_p,
                               float* __restrict__ ws) {
  float cum = 0.f;
  for (int k = 0; k < PHI_K; ++k) {
    float v = phi_log_inc[k];
    float sp = fmaxf(v, 0.f) + log1pf(expf(-fabsf(v)));  // stable softplus
    cum += sp;
    ws[k] = cum;
  }
  float inv = 1.f / (cum + EPSF);
  for (int k = 0; k < PHI_K; ++k) ws[k] *= inv;

  // fused lerp pairs: (value, delta); sentinel at k = PHI_K-1 with delta 0
  for (int k = 0; k < PHI_K - 1; ++k) {
    ws[WS_PAIRS + 2 * k]     = ws[k];
    ws[WS_PAIRS + 2 * k + 1] = ws[k + 1] - ws[k];
  }
  ws[WS_PAIRS + 2 * (PHI_K - 1)]     = ws[PHI_K - 1];
  ws[WS_PAIRS + 2 * (PHI_K - 1) + 1] = 0.f;

  float dmin = 0.f, dmax = 0.f;
  for (int i = 0; i < D_IN; ++i) {
    float l = lambdas[i];
    dmin += fminf(l, 0.f);
    dmax += fmaxf(l, 0.f);
  }
  dmax += (float)(D_OUT - 1);                  // + Q_FACTOR * (D_OUT-1)
  float eta     = eta_p[0];
  float phi_max = 1.f + eta * (float)(D_OUT - 1);
  float h2      = (dmax - dmin) / (float)(PHIU_K - 1);

  ws[100] = phi_max;
  ws[101] = (float)(PHI_K - 1) / phi_max;      // inv knot spacing, stage 1
  ws[102] = dmin;
  ws[103] = 1.f / h2;                          // inv knot spacing, stage 2
  ws[104] = (Phi_coeffs[1] - Phi_coeffs[0]) / h2;                   // left slope
  ws[105] = (Phi_coeffs[PHIU_K - 1] - Phi_coeffs[PHIU_K - 2]) / h2; // right slope
  ws[106] = dmax;
  ws[107] = eta;
}

// ---------------------------------------------------------------------------
// Main kernel: 128 threads (4 wave32s) per block, one lane per output col o.
// x rows staged via CDNA5 async global->LDS DMA (ASYNCcnt); tables in LDS as
// fused (value, delta) pairs with sentinel -> ds_load_b64 + fma, no clamp.
// ---------------------------------------------------------------------------
__global__ __launch_bounds__(D_OUT)
void sprecher_main(const float* __restrict__ x,
                   const float* __restrict__ Phi_coeffs,
                   const float* __restrict__ lambdas,
                   const float* __restrict__ ws,
                   float* __restrict__ out) {
  __shared__ float2 sP1[PHI_K];        // stage-1 (value, delta) pairs + sentinel
  __shared__ float2 sP2[PHIU_K];       // stage-2 (value, delta) pairs + sentinel
  __shared__ float  sLam[D_IN];        // lambdas
  __shared__ float  sX[ROWS * D_IN];   // staged x rows

  const int tid     = threadIdx.x;             // == output index o (0..127)
  const int rowBase = blockIdx.x * ROWS;       // first batch row of this block

  if (tid < PHI_K) {
    sP1[tid] = make_float2(ws[WS_PAIRS + 2 * tid], ws[WS_PAIRS + 2 * tid + 1]);
  }
  if (tid < PHIU_K) {
    float c0 = Phi_coeffs[tid];
    float c1 = (tid < PHIU_K - 1) ? Phi_coeffs[tid + 1] : c0;  // sentinel
    sP2[tid] = make_float2(c0, c1 - c0);
  }
  sLam[tid] = lambdas[tid];

  // --- CDNA5 async global->LDS staging of the ROWS x-rows (ASYNCcnt path) ---
  {
    const float* gbase = x + (size_t)rowBase * D_IN;   // uniform base (SGPR pair)
    unsigned ldsBase = (unsigned)(uintptr_t)(&sX[0]);
    #pragma unroll
    for (int r = 0; r < ROWS; ++r) {
      unsigned voff  = (unsigned)((r * D_IN + tid) * 4);  // per-lane byte offset
      unsigned laddr = ldsBase + voff;                    // per-lane LDS address
      asm volatile(
        "global_load_async_to_lds_b32 %0, %1, %2 offset:0"
        :: "v"(laddr), "v"(voff), "s"(gbase)
        : "memory");
    }
    asm volatile("s_wait_asynccnt 0" ::: "memory");
  }
  __syncthreads();

  // Broadcast scalar parameters (uniform -> scalar loads)
  const float phi_max = ws[100];
  const float inv_h1  = ws[101];
  const float dmin    = ws[102];
  const float inv_h2  = ws[103];
  const float lslope  = ws[104];
  const float rslope  = ws[105];
  const float dmax    = ws[106];
  const float eta     = ws[107];

  const float eo = eta * (float)tid;           // eta * q, constant per lane

  float acc[ROWS];
  #pragma unroll
  for (int r = 0; r < ROWS; ++r) acc[r] = 0.f;

  // s[b,o] = sum_i lambda_i * phi(x[b,i] + eta*o)
  #pragma unroll 2
  for (int i = 0; i < D_IN; ++i) {
    const float lam = sLam[i];                 // broadcast (conflict-free)
    #pragma unroll
    for (int r = 0; r < ROWS; ++r) {
      float z  = sX[r * D_IN + i] + eo;        // broadcast read + per-lane shift
      float zc = fminf(fmaxf(z, 0.f), phi_max);   // v_med3-style clamp
      float t  = zc * inv_h1;                  // uniform grid: [0, 99]
      int  idx = (int)t;                       // floor (t >= 0), in [0,99]
      float fr = __builtin_amdgcn_fractf(t);   // t - floor(t), 0 at sentinel
      float2 p = sP1[idx];                     // one ds_load_b64
      float v  = fmaf(fr, p.y, p.x);           // fused lerp
      v = (z < 0.f)     ? 0.f : v;             // below -> 0
      v = (z > phi_max) ? 1.f : v;             // above -> 1
      acc[r] = fmaf(lam, v, acc[r]);
    }
  }

  // Outer map Phi with linear extrapolation, then store
  #pragma unroll
  for (int r = 0; r < ROWS; ++r) {
    float s  = acc[r] + (float)tid;            // + Q_FACTOR * q
    float sc = fminf(fmaxf(s, dmin), dmax);
    float t  = (sc - dmin) * inv_h2;
    int  idx = (int)t;                         // in [0,99], sentinel-safe
    float fr  = __builtin_amdgcn_fractf(t);
    float2 p  = sP2[idx];
    float res = fmaf(fr, p.y, p.x);
    res = (s < dmin) ? fmaf(lslope, s - dmin, sP2[0].x)         : res;
    res = (s > dmax) ? fmaf(rslope, s - dmax, sP2[PHIU_K - 1].x) : res;
    out[(size_t)(rowBase + r) * D_OUT + tid] = res;
  }
}

// ---------------------------------------------------------------------------
extern "C" void kernel_launch(void* const* d_in, const int* in_sizes, int n_in,
                              void* d_out, int out_size, void* d_ws, size_t ws_size,
                              hipStream_t stream) {
  const float* x        = (const float*)d_in[0];  // [8192,128]
  const float* phi_log  = (const float*)d_in[1];  // [100]
  const float* PhiC     = (const float*)d_in[2];  // [100]
  const float* lam      = (const float*)d_in[3];  // [128]
  const float* eta      = (const float*)d_in[4];  // [1]
  float* ws  = (float*)d_ws;
  float* out = (float*)d_out;

  sprecher_setup<<<1, 1, 0, stream>>>(phi_log, PhiC, lam, eta, ws);
  sprecher_main<<<B_TOTAL / ROWS, D_OUT, 0, stream>>>(x, PhiC, lam, ws, out);
}